// NumericalMarkowitz_86517821215676
// MI455X (gfx1250) — compile-verified
//
#include <hip/hip_runtime.h>
#include <math.h>

typedef __attribute__((ext_vector_type(2))) float v2f;
typedef __attribute__((ext_vector_type(8))) float v8f;
typedef __attribute__((ext_vector_type(4))) unsigned int v4u;
typedef __attribute__((ext_vector_type(8))) int v8i;
typedef __attribute__((ext_vector_type(4))) int v4i;

#define NASSET   128
#define PITCH    129      // odd pitch: conflict-free row-wise LDS reads in the mat-vec
#define N_ITERS  300
#define N_BISECT 50
#define MW       1.0f

#if defined(__has_builtin)
#if __has_builtin(__builtin_amdgcn_tensor_load_to_lds)
#define USE_TDM 1
#endif
#endif

// ---- wave32 shuffle reductions (no barriers) ----
__device__ __forceinline__ float wsum(float x) {
#pragma unroll
  for (int off = 16; off > 0; off >>= 1) x += __shfl_xor(x, off, 32);
  return x;
}
__device__ __forceinline__ float wmin(float x) {
#pragma unroll
  for (int off = 16; off > 0; off >>= 1) x = fminf(x, __shfl_xor(x, off, 32));
  return x;
}
__device__ __forceinline__ float wmax(float x) {
#pragma unroll
  for (int off = 16; off > 0; off >>= 1) x = fmaxf(x, __shfl_xor(x, off, 32));
  return x;
}

extern __shared__ float sM[];   // PITCH*NASSET floats: holds C, later overwritten by Q

__global__ __launch_bounds__(128)
void markowitz_fista(const float* __restrict__ rets,
                     const float* __restrict__ cov,
                     const float* __restrict__ gsq,
                     const float* __restrict__ alphav,
                     float* __restrict__ out)
{
  __shared__ float s_red[4];
  __shared__ __align__(16) float s_y[NASSET];
  __shared__ float s_v[NASSET];

  const int b    = blockIdx.x;
  const int tid  = threadIdx.x;
  const int lane = tid & 31;
  const int wv   = tid >> 5;     // 4 waves
  const int hs   = lane >> 4;    // half-wave select (WMMA frag layout)
  const int lc   = lane & 15;

  // ------------- stage covmat_sqrt[b] (128x128 fp32) into LDS -------------
  const float* __restrict__ Cg = cov + (size_t)b * NASSET * NASSET;

#ifdef USE_TDM
  // Tensor Data Mover: one DMA moves the whole 128x128 tile Global->LDS and
  // inserts 1 DWORD of pad every 128 DWORDs (pad_interval=6, pad_amount=0)
  // => hardware produces the pitch-129 bank-conflict-free layout directly.
  if (wv == 0) {
    const unsigned long long gaddr = (unsigned long long)(uintptr_t)Cg;
    const unsigned int ldsOff = (unsigned int)(uintptr_t)sM;  // low 32 bits = LDS byte offset

    v4u g0;
    g0.x = 1u;                                        // count=1 (valid user D#)
    g0.y = ldsOff;                                    // lds_addr (bytes)
    g0.z = (unsigned int)(gaddr & 0xFFFFFFFFu);       // global_addr[31:0]
    g0.w = (unsigned int)((gaddr >> 32) & 0x01FFFFFFu) | (2u << 30); // addr[56:32] | type=2

    v8i g1;
    g1[0] = (int)((2u << 16) | (1u << 20) | (6u << 22) | (0u << 25));
            // data_size=4B, pad_enable=1, pad_interval=6 (128 DW), pad_amount=0 (1 DW)
    g1[1] = (int)(128u << 16);     // tensor_dim0[15:0]=128   (bits 63:48)
    g1[2] = (int)(128u << 16);     // tensor_dim1[15:0]=128   (bits 95:80)
    g1[3] = (int)(128u << 16);     // tile_dim0=128           (bits 127:112)
    g1[4] = 128;                   // tile_dim1=128, tile_dim2=0
    g1[5] = 128;                   // tensor_dim0_stride[31:0]=128
    g1[6] = (int)(16384u << 16);   // tensor_dim1_stride[15:0]=16384
    g1[7] = 0;                     // tensor_dim1_stride[47:16]=0

    v4i gz4 = {0, 0, 0, 0};
    v8i gz8 = {0, 0, 0, 0, 0, 0, 0, 0};
    __builtin_amdgcn_tensor_load_to_lds(g0, g1, gz4, gz4, gz8, 0);
    __builtin_amdgcn_s_wait_tensorcnt(0);
  }
#else
#pragma unroll 4
  for (int idx = tid; idx < NASSET * NASSET; idx += 128) {
    int r = idx >> 7, c = idx & 127;
    sM[r * PITCH + c] = Cg[idx];
  }
#endif
  __syncthreads();

  const float g  = gsq[b];
  const float g2 = g * g;
  const float aa = fabsf(alphav[b]);

  // ------------- Q = g2 * C^T C + aa*I  via V_WMMA_F32_16X16X4_F32 -------------
  // wave wv owns Q tile-rows {2wv, 2wv+1}; 8 tile-cols each; accumulators in VGPRs.
  v8f acc[2][8];
#pragma unroll
  for (int h = 0; h < 2; ++h)
#pragma unroll
    for (int t = 0; t < 8; ++t)
#pragma unroll
      for (int r = 0; r < 8; ++r)
        acc[h][t][r] = 0.0f;

  const int acol = 32 * wv + lc;           // base column of this wave's A tiles
  for (int k0 = 0; k0 < NASSET; k0 += 4) {
    const int rA = k0 + 2 * hs;            // lanes 0-15: rows k0,k0+1 ; lanes 16-31: k0+2,k0+3
    v2f frag[8];                           // B fragments (also C-columns)
#pragma unroll
    for (int t = 0; t < 8; ++t) {
      frag[t].x = sM[rA * PITCH + 16 * t + lc];
      frag[t].y = sM[(rA + 1) * PITCH + 16 * t + lc];
    }
#pragma unroll
    for (int h = 0; h < 2; ++h) {
      // A fragment loaded directly from LDS (avoids dynamic reg-array indexing)
      v2f A;
      A.x = sM[rA * PITCH + acol + 16 * h];
      A.y = sM[(rA + 1) * PITCH + acol + 16 * h];
#pragma unroll
      for (int t = 0; t < 8; ++t)
        acc[h][t] = __builtin_amdgcn_wmma_f32_16x16x4_f32(
            false, A, false, frag[t], (short)0, acc[h][t], false, false);
    }
  }

  // scale by g2, add |alpha| on the diagonal, accumulate Frobenius norm on the fly
  float frob = 0.0f;
#pragma unroll
  for (int h = 0; h < 2; ++h) {
    const int it = 2 * wv + h;
#pragma unroll
    for (int t = 0; t < 8; ++t) {
#pragma unroll
      for (int r = 0; r < 8; ++r) {
        float q = acc[h][t][r] * g2;
        if (t == it && lc == r + 8 * hs) q += aa;   // diagonal element
        acc[h][t][r] = q;
        frob = fmaf(q, q, frob);
      }
    }
  }
  frob = wsum(frob);
  if (lane == 0) s_red[wv] = frob;
  __syncthreads();                                   // also fences all WMMA LDS reads of C
  frob = s_red[0] + s_red[1] + s_red[2] + s_red[3];
  const float step = 1.0f / (2.0f * sqrtf(frob));

  // write Q over C in LDS (VGPR accum -> LDS, pitch 129 keeps mat-vec conflict-free)
#pragma unroll
  for (int h = 0; h < 2; ++h) {
    const int i0 = 16 * (2 * wv + h) + 8 * hs;
#pragma unroll
    for (int t = 0; t < 8; ++t)
#pragma unroll
      for (int r = 0; r < 8; ++r)
        sM[(i0 + r) * PITCH + 16 * t + lc] = acc[h][t][r];
  }
  __syncthreads();

  // ------------- FISTA: 300 iterations, thread tid owns asset tid -------------
  float w    = 1.0f / (float)NASSET;
  float yv   = w;
  float tcur = 1.0f;
  const float ri = rets[(size_t)b * NASSET + tid];
  const float* __restrict__ qrow = sM + tid * PITCH;

  for (int itn = 0; itn < N_ITERS; ++itn) {
    s_y[tid] = yv;
    __syncthreads();                               // barrier 1 (also WARs s_v reads)

    // grad = 2*Q[i,:]@y - rets[i]   (row-wise LDS reads: conflict-free; y via b128 broadcast)
    float dot = 0.0f;
    const float4* __restrict__ yq = reinterpret_cast<const float4*>(s_y);
#pragma unroll
    for (int k4 = 0; k4 < NASSET / 4; ++k4) {
      float4 yvv = yq[k4];
      dot = fmaf(qrow[4 * k4 + 0], yvv.x, dot);
      dot = fmaf(qrow[4 * k4 + 1], yvv.y, dot);
      dot = fmaf(qrow[4 * k4 + 2], yvv.z, dot);
      dot = fmaf(qrow[4 * k4 + 3], yvv.w, dot);
    }
    const float v = yv - step * (2.0f * dot - ri);

    s_v[tid] = v;
    __syncthreads();                               // barrier 2 (also WARs s_y reads)

    // every wave grabs a redundant full copy of v (4 per lane) -> bisection is
    // pure wave-local shuffle work, ZERO barriers inside the 50-step loop.
    float v4[4];
#pragma unroll
    for (int j = 0; j < 4; ++j) v4[j] = s_v[lane + 32 * j];

    float lo = wmin(fminf(fminf(v4[0], v4[1]), fminf(v4[2], v4[3]))) - MW;
    float hi = wmax(fmaxf(fmaxf(v4[0], v4[1]), fmaxf(v4[2], v4[3])));

    for (int bi = 0; bi < N_BISECT; ++bi) {
      const float tau = 0.5f * (lo + hi);
      float s = 0.0f;
#pragma unroll
      for (int j = 0; j < 4; ++j)
        s += fminf(fmaxf(v4[j] - tau, 0.0f), MW);
      s = wsum(s);
      if (s > 1.0f) lo = tau; else hi = tau;       // uniform across wave
    }
    const float tau0 = 0.5f * (lo + hi);

    // re-derive tau from the fixed active set (wave-local sums)
    float sf = 0.0f, nf = 0.0f, nu = 0.0f;
#pragma unroll
    for (int j = 0; j < 4; ++j) {
      const float z  = v4[j] - tau0;
      const bool  fr = (z > 0.0f) && (z < MW);
      const bool  up = (z >= MW);
      sf += fr ? v4[j] : 0.0f;
      nf += fr ? 1.0f  : 0.0f;
      nu += up ? 1.0f  : 0.0f;
    }
    sf = wsum(sf); nf = wsum(nf); nu = wsum(nu);
    nf = fmaxf(nf, 1.0f);
    const float tau = (sf + MW * nu - 1.0f) / nf;

    // thread's own update from its own v
    const float z  = v - tau0;
    const bool  fr = (z > 0.0f) && (z < MW);
    const bool  up = (z >= MW);
    const float wn = fr ? (v - tau) : (up ? MW : 0.0f);

    const float tn = 0.5f * (1.0f + sqrtf(1.0f + 4.0f * tcur * tcur));
    yv   = wn + ((tcur - 1.0f) / tn) * (wn - w);
    w    = wn;
    tcur = tn;
  }

  out[(size_t)b * NASSET + tid] = w;
}

extern "C" void kernel_launch(void* const* d_in, const int* in_sizes, int n_in,
                              void* d_out, int out_size, void* d_ws, size_t ws_size,
                              hipStream_t stream) {
  const float* rets   = (const float*)d_in[0];
  const float* cov    = (const float*)d_in[1];
  const float* gsq    = (const float*)d_in[2];
  const float* alphav = (const float*)d_in[3];
  float* out = (float*)d_out;

  const int B = in_sizes[2];                       // batch count (1024)
  const size_t smem = (size_t)PITCH * NASSET * sizeof(float);   // ~66 KB dynamic LDS
  markowitz_fista<<<B, 128, smem, stream>>>(rets, cov, gsq, alphav, out);
}